// Beamformer_26766236189144
// MI455X (gfx1250) — compile-verified
//
#include <hip/hip_runtime.h>
#include <hip/hip_bf16.h>

// Problem constants (match reference)
#define BATCH 128
#define NANT  64
#define TSNAP 1024
#define NANG  3600
#define NPEAK 5

typedef float v2f __attribute__((ext_vector_type(2)));
typedef float v8f __attribute__((ext_vector_type(8)));

__device__ __forceinline__ v8f wmma_f32(v2f a, v2f b, v8f c) {
  // D = A(16x4,f32) x B(4x16,f32) + C(16x16,f32); 8-arg form:
  // (neg_a, A, neg_b, B, c_mod, C, reuse_a, reuse_b)
  return __builtin_amdgcn_wmma_f32_16x16x4_f32(false, a, false, b, (short)0, c,
                                               false, false);
}

__device__ __forceinline__ v2f load_pair(const float* p) {
  return *(const v2f*)p;  // 8B-aligned global_load_b64
}

// ---------------------------------------------------------------------------
// Stage 1: R_b = X_b X_b^H / T  (complex, stored as Rre/Rim, row-major 64x64)
// One wave per (b, 16x16 tile). K loop over T=1024 in steps of 4.
// ---------------------------------------------------------------------------
__global__ void __launch_bounds__(32)
cov_kernel(const float* __restrict__ xr, const float* __restrict__ xi,
           float* __restrict__ Rre, float* __restrict__ Rim) {
  const int lane = threadIdx.x;            // 0..31 (wave32)
  const int tile = blockIdx.x & 15;        // 4x4 tiles of 16x16
  const int b    = blockIdx.x >> 4;
  const int ti   = tile >> 2;
  const int tj   = tile & 3;
  const int row  = lane & 15;
  const int kb   = (lane >> 4) << 1;       // 0 for lanes 0-15, 2 for 16-31

  const float* xrA = xr + ((size_t)b * NANT + ti * 16 + row) * TSNAP;
  const float* xiA = xi + ((size_t)b * NANT + ti * 16 + row) * TSNAP;
  const float* xrB = xr + ((size_t)b * NANT + tj * 16 + row) * TSNAP;
  const float* xiB = xi + ((size_t)b * NANT + tj * 16 + row) * TSNAP;

  v8f acc_re = {};  // Σ xr_m xr_n + xi_m xi_n
  v8f acc_p  = {};  // Σ xi_m xr_n
  v8f acc_q  = {};  // Σ xr_m xi_n   (Rim = acc_p - acc_q)

#pragma unroll 4
  for (int k = 0; k < TSNAP; k += 4) {
    v2f ar = load_pair(xrA + k + kb);
    v2f ai = load_pair(xiA + k + kb);
    v2f br = load_pair(xrB + k + kb);
    v2f bi = load_pair(xiB + k + kb);
    acc_re = wmma_f32(ar, br, acc_re);
    acc_re = wmma_f32(ai, bi, acc_re);
    acc_p  = wmma_f32(ai, br, acc_p);
    acc_q  = wmma_f32(ar, bi, acc_q);
  }

  // C/D layout: VGPR r -> M = r (+8 for lanes>=16), N = lane&15
  const int   nn    = tj * 16 + (lane & 15);
  const int   mbase = ti * 16 + ((lane >> 4) << 3);
  const float invT  = 1.0f / (float)TSNAP;
#pragma unroll
  for (int r = 0; r < 8; ++r) {
    size_t off = ((size_t)b * NANT + (mbase + r)) * NANT + nn;
    Rre[off] = acc_re[r] * invT;
    Rim[off] = (acc_p[r] - acc_q[r]) * invT;
  }
}

// ---------------------------------------------------------------------------
// Stage 2: spectrum[b,a] = Re( S(:,a)^H * (R_b * S(:,a)) )
// One wave per (a-tile of 16 angles, batch group of 16). S tile kept in regs.
// ---------------------------------------------------------------------------
__global__ void __launch_bounds__(32)
spectrum_kernel(const float* __restrict__ Rre, const float* __restrict__ Rim,
                const float* __restrict__ sr, const float* __restrict__ si,
                float* __restrict__ spec) {
  const int lane  = threadIdx.x;
  const int atile = blockIdx.x;            // 0..224
  const int bgrp  = blockIdx.y;            // 0..7
  const int a     = atile * 16 + (lane & 15);
  const int kb    = (lane >> 4) << 1;      // 0 or 2
  const int row   = lane & 15;

  // Preload B-operands (b-invariant): S[k][a], k pair at ks*4+kb
  v2f Sr[16], Si[16];
#pragma unroll
  for (int ks = 0; ks < 16; ++ks) {
    const int k0 = ks * 4 + kb;
    v2f tr, tiv;
    tr[0]  = sr[(size_t)k0 * NANG + a];
    tr[1]  = sr[(size_t)(k0 + 1) * NANG + a];
    tiv[0] = si[(size_t)k0 * NANG + a];
    tiv[1] = si[(size_t)(k0 + 1) * NANG + a];
    Sr[ks] = tr;
    Si[ks] = tiv;
  }

  for (int bb = 0; bb < BATCH / 8; ++bb) {
    const int b = bgrp * (BATCH / 8) + bb;
    const float* Rrb = Rre + (size_t)b * NANT * NANT;
    const float* Rib = Rim + (size_t)b * NANT * NANT;
    float part = 0.0f;

#pragma unroll
    for (int mt = 0; mt < 4; ++mt) {
      v8f vrr = {}, vii = {}, vri = {}, vir = {};
      const float* RrRow = Rrb + (mt * 16 + row) * NANT + kb;
      const float* RiRow = Rib + (mt * 16 + row) * NANT + kb;
#pragma unroll
      for (int ks = 0; ks < 16; ++ks) {
        v2f ar = load_pair(RrRow + ks * 4);
        v2f ai = load_pair(RiRow + ks * 4);
        vrr = wmma_f32(ar, Sr[ks], vrr);
        vii = wmma_f32(ai, Si[ks], vii);
        vri = wmma_f32(ar, Si[ks], vri);
        vir = wmma_f32(ai, Sr[ks], vir);
      }
      // Vre = vrr - vii ; Vim = vri + vir
      // Lane holds rows m = mt*16 + r (+8 for lanes>=16) of column a.
      const int mb = mt * 16 + ((lane >> 4) << 3);
#pragma unroll
      for (int r = 0; r < 8; ++r) {
        const float vre = vrr[r] - vii[r];
        const float vim = vri[r] + vir[r];
        const float wr  = sr[(size_t)(mb + r) * NANG + a];
        const float wi  = si[(size_t)(mb + r) * NANG + a];
        part += wr * vre + wi * vim;
      }
    }
    // lane n has m=0..7-half, lane n+16 has m=8..15-half of the same column
    part += __shfl_xor(part, 16, 32);
    if (lane < 16) spec[(size_t)b * NANG + a] = part;
  }
}

// ---------------------------------------------------------------------------
// Stage 3: strict local maxima, top-5 (lowest-index tiebreak), gather angles.
// One 256-thread block per batch row.
// ---------------------------------------------------------------------------
__global__ void __launch_bounds__(256)
peaks_kernel(const float* __restrict__ spec, const float* __restrict__ angles,
             float* __restrict__ out) {
  __shared__ float vals[NANG];
  __shared__ float rmax[8];
  __shared__ int   ridx[8];

  const int b   = blockIdx.x;
  const int tid = threadIdx.x;
  const float* row = spec + (size_t)b * NANG;
  const float NEG = -__builtin_inff();

  for (int i = tid; i < NANG; i += 256) {
    const float v  = row[i];
    const bool  pk = (i > 0) && (i < NANG - 1) && (v > row[i - 1]) && (v > row[i + 1]);
    vals[i] = pk ? v : NEG;
  }
  __syncthreads();

  for (int m = 0; m < NPEAK; ++m) {
    float bv = NEG;
    int   bi = 0x7fffffff;
    for (int i = tid; i < NANG; i += 256) {
      const float v = vals[i];
      if (v > bv || (v == bv && i < bi)) { bv = v; bi = i; }
    }
    // wave32 reduce
    for (int off = 16; off > 0; off >>= 1) {
      const float ov = __shfl_down(bv, off, 32);
      const int   oi = __shfl_down(bi, off, 32);
      if (ov > bv || (ov == bv && oi < bi)) { bv = ov; bi = oi; }
    }
    const int wid = tid >> 5;
    if ((tid & 31) == 0) { rmax[wid] = bv; ridx[wid] = bi; }
    __syncthreads();
    if (tid == 0) {
      float fv = rmax[0];
      int   fi = ridx[0];
      for (int w = 1; w < 8; ++w)
        if (rmax[w] > fv || (rmax[w] == fv && ridx[w] < fi)) { fv = rmax[w]; fi = ridx[w]; }
      out[(size_t)b * NPEAK + m] = angles[fi];
      vals[fi] = NEG;
    }
    __syncthreads();
  }
}

// ---------------------------------------------------------------------------
extern "C" void kernel_launch(void* const* d_in, const int* in_sizes, int n_in,
                              void* d_out, int out_size, void* d_ws, size_t ws_size,
                              hipStream_t stream) {
  (void)in_sizes; (void)n_in; (void)out_size; (void)ws_size;

  const float* xr     = (const float*)d_in[0];  // (B,N,T)
  const float* xi     = (const float*)d_in[1];  // (B,N,T)
  const float* sr     = (const float*)d_in[2];  // (N,A)
  const float* si     = (const float*)d_in[3];  // (N,A)
  const float* angles = (const float*)d_in[4];  // (A)
  // d_in[5] = M (scalar), hardcoded NPEAK=5

  float* ws   = (float*)d_ws;
  float* Rre  = ws;                                  // B*64*64
  float* Rim  = Rre + (size_t)BATCH * NANT * NANT;   // B*64*64
  float* spec = Rim + (size_t)BATCH * NANT * NANT;   // B*A

  cov_kernel<<<dim3(BATCH * 16), dim3(32), 0, stream>>>(xr, xi, Rre, Rim);
  spectrum_kernel<<<dim3(NANG / 16, 8), dim3(32), 0, stream>>>(Rre, Rim, sr, si, spec);
  peaks_kernel<<<dim3(BATCH), dim3(256), 0, stream>>>(spec, angles, (float*)d_out);
}